// CoordsTo_atomNRF_87909390615210
// MI455X (gfx1250) — compile-verified
//
#include <hip/hip_runtime.h>
#include <hip/hip_bf16.h>

// CoordsTo_atomNRF on MI455X (gfx1250, wave32).
// Gram matrix via V_WMMA_F32_16X16X4_F32; branchless NRF epilogue with
// v_rcp_f32 (no IEEE div sequence) and immediate-offset DS weight loads.

typedef __attribute__((ext_vector_type(2))) float v2f;
typedef __attribute__((ext_vector_type(8))) float v8f;

#define NATOMS 64
#define WAVES_PER_BLOCK 8

__global__ __launch_bounds__(256) void
CoordsTo_atomNRF_kernel(const float* __restrict__ coords,
                        const float* __restrict__ atoms_flat,
                        float* __restrict__ out,
                        int nbatch)
{
    // Folded constant: AU2KCALMOLA * (0.5*2/2) / 2 / MAX_ATOM_NRF = * 0.0025
    const float KC = (float)(627.5095 * 0.529177 * 0.0025);

    __shared__ float Wmat[NATOMS * NATOMS];            // 16 KB: symmetric pair weights * KC
    __shared__ float Xsh[WAVES_PER_BLOCK][NATOMS * 3]; // 6 KB: per-wave staged coords
    __shared__ float Nsh[WAVES_PER_BLOCK][NATOMS];     // 2 KB: per-wave |x_i|^2

    const int tid = threadIdx.x;

    // ---- Build 64x64 weight matrix once per block (shared across 8 batches) ----
    for (int idx = tid; idx < NATOMS * NATOMS; idx += 256) {
        int i = idx >> 6;
        int j = idx & 63;
        float w = 0.0f;
        if (i != j) {
            int r = i > j ? i : j;
            int c = i > j ? j : i;
            int p = (r * (r - 1)) / 2 + c;  // strict lower triangle, row-major
            w = atoms_flat[p] * KC;
        }
        Wmat[idx] = w;
    }
    __syncthreads();

    const int wave = tid >> 5;
    const int lane = tid & 31;
    const int batch = blockIdx.x * WAVES_PER_BLOCK + wave;
    if (batch >= nbatch) return;   // wave-uniform; 8192 % 8 == 0 so normally never taken

    const float* __restrict__ cb = coords + (size_t)batch * (NATOMS * 3);
    float* __restrict__ Xw = Xsh[wave];
    float* __restrict__ Nw = Nsh[wave];

    // ---- Stage coords: 192 contiguous floats, coalesced ----
    #pragma unroll
    for (int k = 0; k < 6; ++k)
        Xw[lane + 32 * k] = cb[lane + 32 * k];
    asm volatile("s_wait_dscnt 0" ::: "memory");  // cross-lane LDS visibility in-wave

    // ---- Per-atom squared norms ----
    #pragma unroll
    for (int h = 0; h < 2; ++h) {
        int a = lane + 32 * h;
        float x = Xw[a * 3 + 0];
        float y = Xw[a * 3 + 1];
        float z = Xw[a * 3 + 2];
        Nw[a] = x * x + y * y + z * z;
    }
    asm volatile("s_wait_dscnt 0" ::: "memory");

    // ---- Build WMMA fragments. A(16x4 f32) layout: lanes 0-15 carry K=0,1;
    //      lanes 16-31 carry K=2,3 (K=3 zero-padded). B(4x16) mirrors it, so
    //      one fragment serves as both A and B operand of the Gram product. ----
    const bool hi = lane >= 16;
    const int  l16 = lane & 31 & 15;
    const int  koff = hi ? 2 : 0;
    v2f frag[4];
    #pragma unroll
    for (int t = 0; t < 4; ++t) {
        int a = t * 16 + l16;
        frag[t].x = Xw[a * 3 + koff];
        frag[t].y = hi ? 0.0f : Xw[a * 3 + 1];
    }

    // Runtime part of the Wmat index for this lane: row-half offset + column lane.
    // Everything else (tm*16*64 + tn*16 + v*64) is a compile-time DS immediate.
    const int hrow = hi ? 8 : 0;                 // row offset within a 16-row tile
    const float* __restrict__ Wlane = Wmat + hrow * NATOMS + l16;

    float* __restrict__ ob = out + (size_t)batch * NATOMS;

    // ---- 4x4 tiles of the 64x64 interaction matrix ----
    #pragma unroll
    for (int tm = 0; tm < 4; ++tm) {
        const int rbase = tm * 16 + hrow;        // D layout: VGPR v -> row rbase+v
        float rn[8];
        #pragma unroll
        for (int v = 0; v < 8; ++v)
            rn[v] = Nw[rbase + v];

        float racc[8] = {0.f, 0.f, 0.f, 0.f, 0.f, 0.f, 0.f, 0.f};

        #pragma unroll
        for (int tn = 0; tn < 4; ++tn) {
            v8f d = {};
            // G_tile = frag[tm] (16x4) * frag[tn]-as-B (4x16)
            d = __builtin_amdgcn_wmma_f32_16x16x4_f32(
                    false, frag[tm], false, frag[tn],
                    (short)0, d, false, false);

            const int col = tn * 16 + l16;
            const float cn = Nw[col];
            #pragma unroll
            for (int v = 0; v < 8; ++v) {
                float r2 = rn[v] + cn - 2.0f * d[v];
                // Diagonal only possible in tm==tn tiles (compile-time known);
                // Wmat is 0 there, so neutralizing the denominator suffices.
                float r2s = (tm == tn && (hrow + v) == l16) ? 1.0f : r2;
                float w = Wlane[(tm * 16) * NATOMS + tn * 16 + v * NATOMS];
                racc[v] += w * __builtin_amdgcn_rcpf(r2s);  // v_fmac + v_rcp_f32
            }
        }

        // ---- Row reduce across the 16 lanes of each half-wave ----
        #pragma unroll
        for (int v = 0; v < 8; ++v) {
            float s = racc[v];
            s += __shfl_xor(s, 1, 32);
            s += __shfl_xor(s, 2, 32);
            s += __shfl_xor(s, 4, 32);
            s += __shfl_xor(s, 8, 32);
            racc[v] = s;
        }
        if (l16 == 0) {
            #pragma unroll
            for (int v = 0; v < 8; ++v)
                ob[rbase + v] = racc[v];
        }
    }
}

extern "C" void kernel_launch(void* const* d_in, const int* in_sizes, int n_in,
                              void* d_out, int out_size, void* d_ws, size_t ws_size,
                              hipStream_t stream) {
    const float* coords     = (const float*)d_in[0];
    const float* atoms_flat = (const float*)d_in[1];
    float* out              = (float*)d_out;

    const int nbatch = in_sizes[0] / (NATOMS * 3);   // 8192
    const int blocks = (nbatch + WAVES_PER_BLOCK - 1) / WAVES_PER_BLOCK;

    CoordsTo_atomNRF_kernel<<<blocks, 256, 0, stream>>>(coords, atoms_flat, out, nbatch);
}